// MultiLayerHungarianLoss_65979287601809
// MI455X (gfx1250) — compile-verified
//
#include <hip/hip_runtime.h>
#include <hip/hip_bf16.h>

#define N_PTS 384

typedef float v2f __attribute__((ext_vector_type(2)));
typedef float v4f __attribute__((ext_vector_type(4)));
typedef float v8f __attribute__((ext_vector_type(8)));

// ---------------------------------------------------------------------------
// Zero-init kernel (combined cost accumulator must start at 0 every call).
// ---------------------------------------------------------------------------
__global__ void mlh_zero_kernel(float* __restrict__ p, int n) {
    int i = blockIdx.x * blockDim.x + threadIdx.x;
    if (i < n) p[i] = 0.0f;
}

// ---------------------------------------------------------------------------
// Row squared-norms: one block per row of the (2, N, d) latent tensor.
// norms[r] = sum_k lat[r, k]^2   for r in [0, 2N)
// ---------------------------------------------------------------------------
__global__ void mlh_row_norms_kernel(const float* __restrict__ lat,
                                     float* __restrict__ norms, int d) {
    const int r = blockIdx.x;
    const float* p = lat + (size_t)r * d;
    float s = 0.0f;
    const v4f* p4 = (const v4f*)p;
    const int d4 = d >> 2;
    for (int k = threadIdx.x; k < d4; k += blockDim.x) {
        v4f x = p4[k];
        s += x.x * x.x + x.y * x.y + x.z * x.z + x.w * x.w;
    }
    __shared__ float sm[256];
    sm[threadIdx.x] = s;
    __syncthreads();
    for (int sft = 128; sft > 0; sft >>= 1) {
        if (threadIdx.x < sft) sm[threadIdx.x] += sm[threadIdx.x + sft];
        __syncthreads();
    }
    if (threadIdx.x == 0) norms[r] = sm[0];
}

// ---------------------------------------------------------------------------
// Cost matrix via fp32 WMMA, register-blocked:
//   cost[i][j] = max(sa[i] - 2*dot(A_i,B_j) + sb[j], 0) / d
// One wave owns a 32x32 C tile = 4 x V_WMMA_F32_16X16X4_F32 accumulators.
// Per 8-K step: 4 contiguous b128 fragment loads (K-slot remap trick: dot
// products are K-permutation invariant, A and B use the same slot->memory
// mapping, lane group 0 -> K[k..k+4), group 1 -> K[k+4..k+8)) feed 8 WMMAs,
// interleaved so each accumulator is touched every 4th instruction.
// Block = 256 threads = 8 waves as 2(M) x 4(N) 32-tiles; grid (3, 6).
// ---------------------------------------------------------------------------
__global__ __launch_bounds__(256)
void mlh_cost_wmma_kernel(const float* __restrict__ lat,   // (2, N, d)
                          const float* __restrict__ sa,    // N row norms of A
                          const float* __restrict__ sb,    // N row norms of B
                          float* __restrict__ cost,        // N*N
                          float* __restrict__ combined,    // N*N accumulator
                          int d, float w) {
    const int N = N_PTS;
    const int lane = threadIdx.x & 31;
    const int wave = threadIdx.x >> 5;
    const int tm32 = blockIdx.y * 2 + (wave >> 2);   // 0..11 (M tiles of 32)
    const int tn32 = blockIdx.x * 4 + (wave & 3);    // 0..11 (N tiles of 32)

    const float* A = lat;
    const float* B = lat + (size_t)N * d;

    const int row  = lane & 15;
    const int kgrp = (lane >> 4) * 4;                // lane-group K sub-offset

    const float* aptr0 = A + (size_t)(tm32 * 32 + row) * d + kgrp;
    const float* aptr1 = aptr0 + (size_t)16 * d;
    const float* bptr0 = B + (size_t)(tn32 * 32 + row) * d + kgrp;
    const float* bptr1 = bptr0 + (size_t)16 * d;

    v8f c00 = {0.f, 0.f, 0.f, 0.f, 0.f, 0.f, 0.f, 0.f};
    v8f c01 = c00, c10 = c00, c11 = c00;

#pragma unroll 2
    for (int k = 0; k < d; k += 8) {
        v4f a0 = *(const v4f*)(aptr0 + k);
        v4f a1 = *(const v4f*)(aptr1 + k);
        v4f b0 = *(const v4f*)(bptr0 + k);
        v4f b1 = *(const v4f*)(bptr1 + k);
        v2f a0l = __builtin_shufflevector(a0, a0, 0, 1);
        v2f a0h = __builtin_shufflevector(a0, a0, 2, 3);
        v2f a1l = __builtin_shufflevector(a1, a1, 0, 1);
        v2f a1h = __builtin_shufflevector(a1, a1, 2, 3);
        v2f b0l = __builtin_shufflevector(b0, b0, 0, 1);
        v2f b0h = __builtin_shufflevector(b0, b0, 2, 3);
        v2f b1l = __builtin_shufflevector(b1, b1, 0, 1);
        v2f b1h = __builtin_shufflevector(b1, b1, 2, 3);
        c00 = __builtin_amdgcn_wmma_f32_16x16x4_f32(false, a0l, false, b0l, (short)0, c00, false, false);
        c01 = __builtin_amdgcn_wmma_f32_16x16x4_f32(false, a0l, false, b1l, (short)0, c01, false, false);
        c10 = __builtin_amdgcn_wmma_f32_16x16x4_f32(false, a1l, false, b0l, (short)0, c10, false, false);
        c11 = __builtin_amdgcn_wmma_f32_16x16x4_f32(false, a1l, false, b1l, (short)0, c11, false, false);
        c00 = __builtin_amdgcn_wmma_f32_16x16x4_f32(false, a0h, false, b0h, (short)0, c00, false, false);
        c01 = __builtin_amdgcn_wmma_f32_16x16x4_f32(false, a0h, false, b1h, (short)0, c01, false, false);
        c10 = __builtin_amdgcn_wmma_f32_16x16x4_f32(false, a1h, false, b0h, (short)0, c10, false, false);
        c11 = __builtin_amdgcn_wmma_f32_16x16x4_f32(false, a1h, false, b1h, (short)0, c11, false, false);
    }

    // C layout per accumulator: VGPR r -> M = r + (lane>=16 ? 8 : 0), N = lane&15.
    const float inv_d = 1.0f / (float)d;
    const int mofs = (lane >> 4) * 8;
    const int jlane = lane & 15;
    const v8f* accs[4] = {&c00, &c01, &c10, &c11};
#pragma unroll
    for (int sub = 0; sub < 4; ++sub) {
        const int cm = sub >> 1;          // {c00,c01}->m sub 0; {c10,c11}->m sub 1
        const int cn = sub & 1;
        const v8f& c = *accs[sub];
        const int jj = tn32 * 32 + cn * 16 + jlane;
        const float sbj = sb[jj];
#pragma unroll
        for (int r = 0; r < 8; ++r) {
            const int ii = tm32 * 32 + cm * 16 + mofs + r;
            float sq = sa[ii] - 2.0f * c[r] + sbj;
            float val = fmaxf(sq, 0.0f) * inv_d;
            cost[(size_t)ii * N + jj] = val;
            combined[(size_t)ii * N + jj] += w * val;  // layer kernels serialize on stream
        }
    }
}

// ---------------------------------------------------------------------------
// Jonker-Volgenant LAP (exactly the reference algorithm), one block of
// n=384 threads. Column j (1..n) is owned by thread j-1. Per augmenting step:
// parallel minv/way update over unused columns, LDS tree arg-min (strict '<'
// keeps the left operand on ties; lowest index not guaranteed), parallel dual
// update, serial augment by thread 0.
// ---------------------------------------------------------------------------
__global__ __launch_bounds__(N_PTS)
void mlh_lap_kernel(const float* __restrict__ cost, int* __restrict__ col_of_row) {
    const int n = N_PTS;
    __shared__ float u[N_PTS + 1], v[N_PTS + 1], minv[N_PTS + 1];
    __shared__ int   p[N_PTS + 1], way[N_PTS + 1], used[N_PTS + 1];
    __shared__ float red_val[N_PTS];
    __shared__ int   red_idx[N_PTS];
    __shared__ int   s_j0, s_i0, s_j1, s_done;
    __shared__ float s_delta;

    const int t = threadIdx.x;   // 0..n-1
    const float INF = 1e18f;

    for (int j = t; j <= n; j += n) { u[j] = 0.f; v[j] = 0.f; p[j] = 0; way[j] = 0; }
    __syncthreads();

    for (int i = 1; i <= n; ++i) {
        if (t == 0) { p[0] = i; s_j0 = 0; }
        for (int j = t; j <= n; j += n) { minv[j] = INF; used[j] = 0; }
        __syncthreads();

        while (true) {
            if (t == 0) { used[s_j0] = 1; s_i0 = p[s_j0]; }
            __syncthreads();
            const int i0 = s_i0;
            const int j0 = s_j0;

            const int j = t + 1;               // columns 1..n
            float myv = INF;
            if (!used[j]) {
                float cur = cost[(size_t)(i0 - 1) * n + (j - 1)] - u[i0] - v[j];
                if (cur < minv[j]) { minv[j] = cur; way[j] = j0; }
                myv = minv[j];
            }
            red_val[t] = myv;
            red_idx[t] = j;
            __syncthreads();
            for (int s = 256; s > 0; s >>= 1) {
                if (t < s && t + s < n) {
                    if (red_val[t + s] < red_val[t]) {
                        red_val[t] = red_val[t + s];
                        red_idx[t] = red_idx[t + s];
                    }
                }
                __syncthreads();
            }
            if (t == 0) { s_delta = red_val[0]; s_j1 = red_idx[0]; }
            __syncthreads();
            const float delta = s_delta;

            // dual update; p[] values over used columns are distinct -> no races
            for (int jj = t; jj <= n; jj += n) {
                if (used[jj]) { u[p[jj]] += delta; v[jj] -= delta; }
                else          { minv[jj] -= delta; }
            }
            __syncthreads();
            if (t == 0) { s_j0 = s_j1; s_done = (p[s_j1] == 0); }
            __syncthreads();
            if (s_done) break;
        }

        if (t == 0) {
            int j0 = s_j0;
            while (j0) { int j1 = way[j0]; p[j0] = p[j1]; j0 = j1; }
        }
        __syncthreads();
    }

    const int j = t + 1;
    if (p[j] >= 1) col_of_row[p[j] - 1] = j - 1;
}

// ---------------------------------------------------------------------------
// total = sum_l w_l * 2*mismatch_l/N ; out = [total, arange(N), col_ind] (float)
// colind buffer: 5 solves -> layers 0..3 then combined at slot 4.
// ---------------------------------------------------------------------------
__global__ __launch_bounds__(N_PTS)
void mlh_finalize_kernel(const int* __restrict__ colind,
                         const int* __restrict__ invA,
                         const int* __restrict__ invB,
                         float* __restrict__ out) {
    const int n = N_PTS;
    __shared__ int ideal_col[N_PTS];
    __shared__ int mm[4];
    const int t = threadIdx.x;
    if (t < 4) mm[t] = 0;
    __syncthreads();
    ideal_col[invA[t]] = invB[t];
    __syncthreads();
#pragma unroll
    for (int l = 0; l < 4; ++l) {
        if (colind[l * n + t] != ideal_col[t]) atomicAdd(&mm[l], 1);
    }
    __syncthreads();
    if (t == 0) {
        const float wts[4] = {1.0f, 0.5f, 0.25f, 0.125f};
        float total = 0.0f;
#pragma unroll
        for (int l = 0; l < 4; ++l) total += wts[l] * 2.0f * (float)mm[l] / (float)n;
        out[0] = total;
    }
    out[1 + t]     = (float)t;                    // row_ind
    out[1 + n + t] = (float)colind[4 * n + t];    // col_ind from combined LAP
}

// ---------------------------------------------------------------------------
extern "C" void kernel_launch(void* const* d_in, const int* in_sizes, int n_in,
                              void* d_out, int out_size, void* d_ws, size_t ws_size,
                              hipStream_t stream) {
    const int N = N_PTS;
    const int dims[4] = {16384, 16384, 8192, 8192};
    const float wts[4] = {1.0f, 0.5f, 0.25f, 0.125f};

    // Workspace layout (floats):
    //   [0, 4*N*N)        : per-layer cost matrices
    //   [4*N*N, 5*N*N)    : combined cost
    //   [5*N*N, 5*N*N+8N) : row norms (4 layers x 2N)
    //   then 5*N ints     : col_of_row for 4 layers + combined
    float* ws       = (float*)d_ws;
    float* costbuf  = ws;
    float* combined = ws + (size_t)4 * N * N;
    float* norms    = ws + (size_t)5 * N * N;
    int*   colind   = (int*)(ws + (size_t)5 * N * N + 8 * N);

    mlh_zero_kernel<<<(N * N + 255) / 256, 256, 0, stream>>>(combined, N * N);

    for (int l = 0; l < 4; ++l) {
        const float* lat = (const float*)d_in[l];
        mlh_row_norms_kernel<<<2 * N, 256, 0, stream>>>(lat, norms + l * 2 * N, dims[l]);
    }

    for (int l = 0; l < 4; ++l) {
        const float* lat = (const float*)d_in[l];
        dim3 grid(3, 6);   // x: 3 blocks x 4 waves = 12 N-tiles(32); y: 6 x 2 = 12 M-tiles(32)
        mlh_cost_wmma_kernel<<<grid, 256, 0, stream>>>(
            lat, norms + l * 2 * N, norms + l * 2 * N + N,
            costbuf + (size_t)l * N * N, combined, dims[l], wts[l]);
    }

    for (int l = 0; l < 4; ++l)
        mlh_lap_kernel<<<1, N, 0, stream>>>(costbuf + (size_t)l * N * N, colind + l * N);
    mlh_lap_kernel<<<1, N, 0, stream>>>(combined, colind + 4 * N);

    mlh_finalize_kernel<<<1, N, 0, stream>>>(colind,
                                             (const int*)d_in[4],
                                             (const int*)d_in[5],
                                             (float*)d_out);
}